// RegPool_9208409882645
// MI455X (gfx1250) — compile-verified
//
#include <hip/hip_runtime.h>

typedef __attribute__((ext_vector_type(2))) float v2f;
typedef __attribute__((ext_vector_type(8))) float v8f;

#define PLEN 20
#define HDIM 1024

// ---------------------------------------------------------------------------
// CDNA5 async global->LDS DMA (ASYNCcnt path, no staging VGPRs).
// GVS addressing: saddr = SGPR64 base, vaddr = per-lane u32 byte offset,
// vdst = per-lane LDS byte address (32-bit addrspace(3) pointer).
// ---------------------------------------------------------------------------
__device__ __forceinline__
void async_copy_b128(float* lds_dst, const float* gbase, unsigned voff_bytes) {
#if defined(__gfx1250__)
    asm volatile("global_load_async_to_lds_b128 %0, %1, %2"
                 :
                 : "v"((__attribute__((address_space(3))) float*)lds_dst),
                   "v"(voff_bytes),
                   "s"(gbase)
                 : "memory");
#else
    *(float4*)lds_dst = *(const float4*)((const char*)gbase + voff_bytes);
#endif
}

__device__ __forceinline__ void wait_async0() {
#if defined(__gfx1250__)
    asm volatile("s_wait_asynccnt 0x0" ::: "memory");
#endif
}

// ---------------------------------------------------------------------------
// Kernel 1: pooled[m][h] = (1/len[m]) * sum_p language[m][p][h]
// Pure streaming pass: 168 MB read, 8 MB written -> HBM-bound (~7.5 us).
// ---------------------------------------------------------------------------
__global__ __launch_bounds__(256)
void pool_kernel(const float* __restrict__ lang,
                 const int* __restrict__ plen,
                 float* __restrict__ pooled) {
    const int m = blockIdx.x;
    const float4* base = (const float4*)(lang + (size_t)m * (PLEN * HDIM));
    const float inv = 1.0f / (float)plen[m];
    const int t = threadIdx.x;              // 0..255 -> one float4 of H
    float4 s = make_float4(0.f, 0.f, 0.f, 0.f);
#pragma unroll
    for (int p = 0; p < PLEN; ++p) {
        float4 v = base[p * (HDIM / 4) + t];
        s.x += v.x; s.y += v.y; s.z += v.z; s.w += v.w;
    }
    float4 r;
    r.x = s.x * inv; r.y = s.y * inv; r.z = s.z * inv; r.w = s.w * inv;
    ((float4*)(pooled + (size_t)m * HDIM))[t] = r;
}

// ---------------------------------------------------------------------------
// Kernel 2: C[M,N] = A[M,K] @ W[K,N] + bias[N], all f32, via
// V_WMMA_F32_16X16X4_F32 (full-precision matrix pipe, 2048 FLOP/instr).
//
// Block tile 64(M) x 128(N), BK=32, 256 threads = 8 waves (2Mx4N), each wave
// owns a 32x32 C tile (4 v8f accumulators, 4 WMMAs / k-step from 2x2 frags).
//
// Staging: double-buffered LDS + GLOBAL_LOAD_ASYNC_TO_LDS_B128. Loads for
// tile t+1 are issued before computing tile t; one s_wait_asynccnt 0 +
// barrier per iteration, fully overlapped with the 32-WMMA compute phase.
//
// LDS padding: A stride 36 floats (16-row column gather hits 16 distinct
// banks), B stride 136 floats (half-wave rows on disjoint bank sets); both
// give 16B-aligned rows for the B128 DMA.
// ---------------------------------------------------------------------------
__global__ __launch_bounds__(256)
void gemm_f32_wmma(const float* __restrict__ A,
                   const float* __restrict__ W,
                   const float* __restrict__ bias,
                   float* __restrict__ C,
                   int M, int N, int K) {
    constexpr int BM = 64, BN = 128, BK = 32;
    constexpr int LDA = 36;    // padded LDS row stride (floats) for A tile
    constexpr int LDB = 136;   // padded LDS row stride (floats) for B tile
    __shared__ float sA[2][BM * LDA];   // 2 x 9216 B
    __shared__ float sB[2][BK * LDB];   // 2 x 17408 B

    const int t    = threadIdx.x;
    const int lane = t & 31;
    const int wave = t >> 5;          // 0..7
    const int waveM = wave & 1;       // 0..1
    const int waveN = wave >> 1;      // 0..3
    const int r    = lane & 15;       // row/col within fragment
    const int half = lane >> 4;       // lane half -> K sub-select
    const int kb   = half * 2;

    const int rowBase = blockIdx.y * BM;
    const int colBase = blockIdx.x * BN;
    const int wRow = waveM * 32;      // wave tile origin inside block tile
    const int wCol = waveN * 32;

    // --- per-thread DMA coordinates (constant across K loop) ---
    // A tile: 64x32 = 512 float4, 2 per thread
    const int ar0 = t >> 3;                 // 0..31
    const int ar1 = ar0 + 32;               // 32..63
    const int ac  = (t & 7) << 2;           // 0,4,...,28
    const unsigned aBase0 = (unsigned)((rowBase + ar0) * K + ac) * 4u;
    const unsigned aBase1 = (unsigned)((rowBase + ar1) * K + ac) * 4u;
    // B tile: 32x128 = 1024 float4, 4 per thread (rows wave, wave+8, ...)
    const int bRow = t >> 5;                // 0..7
    const int bCol = (t & 31) << 2;         // 0,4,...,124
    unsigned bBase[4];
#pragma unroll
    for (int i = 0; i < 4; ++i)
        bBase[i] = (unsigned)((bRow + i * 8) * N + colBase + bCol) * 4u;
    const unsigned bStep = (unsigned)N * 4u;   // bytes per K row of W

    auto stage = [&](int buf, int kt) {
        const unsigned ka = (unsigned)kt * 4u;
        async_copy_b128(&sA[buf][ar0 * LDA + ac], A, aBase0 + ka);
        async_copy_b128(&sA[buf][ar1 * LDA + ac], A, aBase1 + ka);
        const unsigned kw = (unsigned)kt * bStep;
#pragma unroll
        for (int i = 0; i < 4; ++i)
            async_copy_b128(&sB[buf][(bRow + i * 8) * LDB + bCol], W,
                            bBase[i] + kw);
    };

    v8f acc00 = {}, acc01 = {}, acc10 = {}, acc11 = {};

    // prologue: DMA tile 0
    stage(0, 0);
    wait_async0();
    __syncthreads();

    int cur = 0;
    for (int kt = 0; kt < K; kt += BK) {
        const int nxt = cur ^ 1;
        if (kt + BK < K) stage(nxt, kt + BK);   // issue DMA, no wait

        const float* pa = sA[cur];
        const float* pb = sB[cur];
#pragma unroll
        for (int k0 = 0; k0 < BK; k0 += 4) {
            v2f a0, a1, b0, b1;
            a0.x = pa[(wRow + r) * LDA + k0 + kb];
            a0.y = pa[(wRow + r) * LDA + k0 + kb + 1];
            a1.x = pa[(wRow + 16 + r) * LDA + k0 + kb];
            a1.y = pa[(wRow + 16 + r) * LDA + k0 + kb + 1];
            b0.x = pb[(k0 + kb) * LDB + wCol + r];
            b0.y = pb[(k0 + kb + 1) * LDB + wCol + r];
            b1.x = pb[(k0 + kb) * LDB + wCol + 16 + r];
            b1.y = pb[(k0 + kb + 1) * LDB + wCol + 16 + r];
            acc00 = __builtin_amdgcn_wmma_f32_16x16x4_f32(
                false, a0, false, b0, (short)0, acc00, false, false);
            acc01 = __builtin_amdgcn_wmma_f32_16x16x4_f32(
                false, a0, false, b1, (short)0, acc01, false, false);
            acc10 = __builtin_amdgcn_wmma_f32_16x16x4_f32(
                false, a1, false, b0, (short)0, acc10, false, false);
            acc11 = __builtin_amdgcn_wmma_f32_16x16x4_f32(
                false, a1, false, b1, (short)0, acc11, false, false);
        }
        wait_async0();        // DMA for next tile has had whole compute phase
        __syncthreads();
        cur = nxt;
    }

    // --- epilogue: C layout -> global, + bias ---
    const int cRow0 = rowBase + wRow;        // acc00 / acc01
    const int cRow1 = rowBase + wRow + 16;   // acc10 / acc11
    const int col0  = colBase + wCol + r;
    const int col1  = colBase + wCol + 16 + r;
    const float bias0 = bias[col0];
    const float bias1 = bias[col1];
#pragma unroll
    for (int i = 0; i < 8; ++i) {
        const int m0 = cRow0 + i + half * 8;
        const int m1 = cRow1 + i + half * 8;
        C[(size_t)m0 * N + col0] = acc00[i] + bias0;
        C[(size_t)m0 * N + col1] = acc01[i] + bias1;
        C[(size_t)m1 * N + col0] = acc10[i] + bias0;
        C[(size_t)m1 * N + col1] = acc11[i] + bias1;
    }
}

// ---------------------------------------------------------------------------
// Driver: pool -> language GEMM (out[0:2M)) -> vision GEMM (out[2M:4M))
// Output order matches reference tuple: (language_map, visual_map).
// ---------------------------------------------------------------------------
extern "C" void kernel_launch(void* const* d_in, const int* in_sizes, int n_in,
                              void* d_out, int out_size, void* d_ws, size_t ws_size,
                              hipStream_t stream) {
    const float* vision   = (const float*)d_in[0];   // [B, NB, FD]
    const float* language = (const float*)d_in[1];   // [B, NB, PL, H]
    const int*   plen     = (const int*)d_in[2];     // [B, NB]
    const float* Wv       = (const float*)d_in[3];   // [FD, H]
    const float* bv       = (const float*)d_in[4];   // [H]
    const float* Wl       = (const float*)d_in[5];   // [H, H]
    const float* bl       = (const float*)d_in[6];   // [H]
    float* out = (float*)d_out;

    const int M  = 32 * 64;   // B*NB = 2048
    const int H  = 1024;
    const int FD = 4096;

    float* pooled = (float*)d_ws;   // 2048*1024 f32 = 8 MB scratch

    pool_kernel<<<M, 256, 0, stream>>>(language, plen, pooled);

    dim3 block(256);
    dim3 grid(H / 128, M / 64);     // (8, 32) = 256 blocks per GEMM
    // language_map = pooled @ Wl + bl
    gemm_f32_wmma<<<grid, block, 0, stream>>>(pooled, Wl, bl, out, M, H, H);
    // visual_map = vision @ Wv + bv
    gemm_f32_wmma<<<grid, block, 0, stream>>>(vision, Wv, bv,
                                              out + (size_t)M * H, M, H, FD);
}